// SparseMaxPoolTestTorch_77541339562454
// MI455X (gfx1250) — compile-verified
//
#include <hip/hip_runtime.h>
#include <stdint.h>

// Sparse max-pool (two fused 3/2/1 pools == one 7-wide stride-4 window).
// out[b,oz,oy,ox,c] = max feats over active points with coords in
// [4o-3, 4o+3]^3, else 0.  Output: (2,24,24,24,64) f32.

#define NCH 64
#define OD 24

static __device__ __forceinline__ int imin(int a, int b) { return a < b ? a : b; }

// ---- Kernel 1: fill output with -inf bit pattern (0xFF800000) ----
__global__ void k_init_neginf(uint32_t* __restrict__ out, int n) {
  int i = (blockIdx.x * blockDim.x + threadIdx.x) * 4;
  if (i + 3 < n) {
    uint4 v = make_uint4(0xFF800000u, 0xFF800000u, 0xFF800000u, 0xFF800000u);
    *(uint4*)(out + i) = v;
  } else {
    for (int j = i; j < n; ++j) out[j] = 0xFF800000u;
  }
}

// ---- Kernel 2: scatter-max. One wave32 per point, 2 channels per lane. ----
__global__ __launch_bounds__(256) void k_scatter_max(
    const float* __restrict__ feats, const int* __restrict__ coors,
    float* __restrict__ out, int npts) {
  // LDS staging: 8 feature rows (8*64 f32 = 2 KB) + 8 coor rows (128 B),
  // filled with CDNA5 async global->LDS copies (ASYNCcnt path).
  __shared__ __align__(16) float sfeat[8 * NCH];
  __shared__ __align__(16) int   scoor[8 * 4];

  const int t  = threadIdx.x;
  const int p0 = blockIdx.x * 8;
  const int nb = imin(8, npts - p0);

  // Stage features: nb rows * 256B -> 16B per lane for lanes [0, nb*16).
  if (t < nb * 16) {
    uint32_t lo = (uint32_t)(uintptr_t)sfeat + (uint32_t)(t * 16);
    uint64_t ga = (uint64_t)(uintptr_t)(feats + (size_t)p0 * NCH) + (uint64_t)(t * 16);
    asm volatile("global_load_async_to_lds_b128 %0, %1, off"
                 :: "v"(lo), "v"(ga) : "memory");
  }
  // Stage coors: 16B per point for lanes [128, 128+nb).
  if (t >= 128 && t < 128 + nb) {
    const int w = t - 128;
    uint32_t lo = (uint32_t)(uintptr_t)scoor + (uint32_t)(w * 16);
    uint64_t ga = (uint64_t)(uintptr_t)(coors + (size_t)(p0 + w) * 4);
    asm volatile("global_load_async_to_lds_b128 %0, %1, off"
                 :: "v"(lo), "v"(ga) : "memory");
  }
  // Each issuing wave drains its own ASYNCcnt, then workgroup barrier makes
  // the LDS writes visible to all waves.
  asm volatile("s_wait_asynccnt 0" ::: "memory");
  __syncthreads();

  const int w = t >> 5;   // wave id == point slot (uniform per wave)
  const int l = t & 31;   // lane -> channels 2l, 2l+1
  if (w >= nb) return;

  const int b = __builtin_amdgcn_readfirstlane(scoor[w * 4 + 0]);
  const int z = __builtin_amdgcn_readfirstlane(scoor[w * 4 + 1]);
  const int y = __builtin_amdgcn_readfirstlane(scoor[w * 4 + 2]);
  const int x = __builtin_amdgcn_readfirstlane(scoor[w * 4 + 3]);

  const float2 v = ((const float2*)sfeat)[w * 32 + l];

  // Window o covers z in [4o-3, 4o+3]  =>  o in [ceil((z-3)/4), floor((z+3)/4)]
  //   lo = z>>2 (always in [0,23]),  hi = min(23, (z+3)>>2)
  const int lz = z >> 2, hz = imin(OD - 1, (z + 3) >> 2);
  const int ly = y >> 2, hy = imin(OD - 1, (y + 3) >> 2);
  const int lx = x >> 2, hx = imin(OD - 1, (x + 3) >> 2);

  for (int oz = lz; oz <= hz; ++oz)
    for (int oy = ly; oy <= hy; ++oy)
      for (int ox = lx; ox <= hx; ++ox) {
        float* p = out +
            ((size_t)(((b * OD + oz) * OD + oy) * OD + ox) * NCH + (l << 1));
        // gfx12 float max atomic, no return (STOREcnt), device scope: lands
        // in L2 where the whole 7MB output lives.
        asm volatile("global_atomic_max_num_f32 %0, %1, off scope:SCOPE_DEV"
                     :: "v"(p), "v"(v.x) : "memory");
        asm volatile("global_atomic_max_num_f32 %0, %1, off scope:SCOPE_DEV"
                     :: "v"(p + 1), "v"(v.y) : "memory");
      }
}

// ---- Kernel 3: untouched cells (exact -inf bit pattern) -> 0 ----
__global__ void k_finalize(uint32_t* __restrict__ out, int n) {
  int i = (blockIdx.x * blockDim.x + threadIdx.x) * 4;
  if (i + 3 < n) {
    uint4 v = *(uint4*)(out + i);
    v.x = (v.x == 0xFF800000u) ? 0u : v.x;
    v.y = (v.y == 0xFF800000u) ? 0u : v.y;
    v.z = (v.z == 0xFF800000u) ? 0u : v.z;
    v.w = (v.w == 0xFF800000u) ? 0u : v.w;
    *(uint4*)(out + i) = v;
  } else {
    for (int j = i; j < n; ++j)
      if (out[j] == 0xFF800000u) out[j] = 0u;
  }
}

extern "C" void kernel_launch(void* const* d_in, const int* in_sizes, int n_in,
                              void* d_out, int out_size, void* d_ws, size_t ws_size,
                              hipStream_t stream) {
  const float* feats = (const float*)d_in[0];   // (N, 64) f32
  const int*   coors = (const int*)d_in[1];     // (N, 4)  i32 [b,z,y,x]
  float* out = (float*)d_out;                   // (2,24,24,24,64) f32

  const int npts = in_sizes[1] / 4;
  const int nq   = (out_size + 3) / 4;          // uint4 chunks

  k_init_neginf<<<(nq + 255) / 256, 256, 0, stream>>>((uint32_t*)out, out_size);
  k_scatter_max<<<(npts + 7) / 8, 256, 0, stream>>>(feats, coors, out, npts);
  k_finalize<<<(nq + 255) / 256, 256, 0, stream>>>((uint32_t*)out, out_size);
}